// SPR_RGCN_88648124990635
// MI455X (gfx1250) — compile-verified
//
#include <hip/hip_runtime.h>
#include <hip/hip_bf16.h>

typedef float v2f __attribute__((ext_vector_type(2)));
typedef float v8f __attribute__((ext_vector_type(8)));

// ---------------------------------------------------------------------------
// 0) zero a float region (grid-stride)
// ---------------------------------------------------------------------------
__global__ void k_zero(float* __restrict__ p, size_t n) {
    size_t i = (size_t)blockIdx.x * blockDim.x + threadIdx.x;
    size_t stride = (size_t)gridDim.x * blockDim.x;
    for (; i < n; i += stride) p[i] = 0.0f;
}

// ---------------------------------------------------------------------------
// 1) embedding concat + pre-linear + relu: x0[n,32]
//    one thread per (node, out-col); 32 cols -> lanes of a wave share a node
// ---------------------------------------------------------------------------
__global__ __launch_bounds__(256)
void k_embed_pre(const int* __restrict__ node_feats,   // [N,2]
                 const float* __restrict__ shape_emb,  // [16,8]
                 const float* __restrict__ color_emb,  // [8,8]
                 const float* __restrict__ pre_w,      // [16,32]
                 const float* __restrict__ pre_b,      // [32]
                 float* __restrict__ x0,               // [N,32]
                 int n_nodes) {
    int idx = blockIdx.x * blockDim.x + threadIdx.x;
    int node = idx >> 5;
    int col  = idx & 31;
    if (node >= n_nodes) return;
    int s = node_feats[2 * node + 0];
    int c = node_feats[2 * node + 1];
    float acc = pre_b[col];
    #pragma unroll
    for (int k = 0; k < 8; ++k)
        acc += shape_emb[s * 8 + k] * pre_w[k * 32 + col];
    #pragma unroll
    for (int k = 0; k < 8; ++k)
        acc += color_emb[c * 8 + k] * pre_w[(8 + k) * 32 + col];
    x0[(size_t)node * 32 + col] = fmaxf(acc, 0.0f);
}

// ---------------------------------------------------------------------------
// 2) edge scatter: agg[dst][rel][ch] += x[src][ch] ; cnt[dst][rel] += 1
//    C threads per edge (C = feature width, power of two)
// ---------------------------------------------------------------------------
template <int C, bool ADD_CNT>
__global__ __launch_bounds__(256)
void k_scatter(const int* __restrict__ edge_index,  // [2,E]
               const int* __restrict__ edge_type,   // [E]
               const float* __restrict__ xin,       // [N,C]
               float* __restrict__ agg,             // [N,3,C]
               float* __restrict__ cnt,             // [N,3]
               int n_edges) {
    long long idx = (long long)blockIdx.x * blockDim.x + threadIdx.x;
    int e  = (int)(idx / C);
    int ch = (int)(idx % C);
    if (e >= n_edges) return;
    int s = edge_index[e];
    int d = edge_index[n_edges + e];
    int r = edge_type[e];
    atomicAdd(&agg[((size_t)d * 3 + r) * C + ch], xin[(size_t)s * C + ch]);
    if (ADD_CNT && ch == 0) atomicAdd(&cnt[(size_t)d * 3 + r], 1.0f);
}

// ---------------------------------------------------------------------------
// 3) normalize: agg[g][ch] /= max(cnt[g],1)
// ---------------------------------------------------------------------------
template <int C>
__global__ __launch_bounds__(256)
void k_norm(float* __restrict__ agg, const float* __restrict__ cnt, int n_groups) {
    int i = blockIdx.x * blockDim.x + threadIdx.x;
    if (i >= n_groups * C) return;
    int g = i / C;
    agg[i] = agg[i] / fmaxf(cnt[g], 1.0f);
}

// ---------------------------------------------------------------------------
// 4) fused node transform via WMMA (fp32 16x16x4):
//    out = relu( X@root + sum_r Agg_r@rel_r + bias )
//    one wave per 16-node tile; 4 v8f accumulators cover 16x64 output
// ---------------------------------------------------------------------------
template <int K>
__global__ __launch_bounds__(256)
void k_transform(const float* __restrict__ xin,   // [N,K]
                 const float* __restrict__ agg,   // [N,3,K] (normalized)
                 const float* __restrict__ root,  // [K,64]
                 const float* __restrict__ rel,   // [3,K,64]
                 const float* __restrict__ bias,  // [64]
                 float* __restrict__ xout,        // [N,64]
                 int n_nodes) {
    const int wave = threadIdx.x >> 5;
    const int lane = threadIdx.x & 31;
    const int tile = blockIdx.x * (blockDim.x >> 5) + wave;
    const int ntiles = (n_nodes + 15) >> 4;
    if (tile >= ntiles) return;              // wave-uniform: EXEC stays all-ones
    const int m0   = tile << 4;
    const int lrow = lane & 15;
    const int half = lane >> 4;

    // init accumulators with bias (every D element in lane L has N-col = n0+lrow)
    v8f c0, c1, c2, c3;
    {
        float b0 = bias[lrow], b1 = bias[16 + lrow], b2 = bias[32 + lrow], b3 = bias[48 + lrow];
        #pragma unroll
        for (int i = 0; i < 8; ++i) { c0[i] = b0; c1[i] = b1; c2[i] = b2; c3[i] = b3; }
    }

    // clamp A row for tail tiles (loads only; stores are predicated)
    const int arow_idx = min(m0 + lrow, n_nodes - 1);

    #pragma unroll
    for (int p = 0; p < 4; ++p) {
        const float* Ap;
        const float* Bp;
        int astride;
        if (p == 0) { Ap = xin;  astride = K;     Bp = root; }
        else        { Ap = agg + (size_t)(p - 1) * K; astride = 3 * K; Bp = rel + (size_t)(p - 1) * K * 64; }
        const float* arow = Ap + (size_t)arow_idx * astride + 2 * half;

        for (int k0 = 0; k0 < K; k0 += 4) {
            v2f a;
            a.x = arow[k0];
            a.y = arow[k0 + 1];
            const float* brow = Bp + (size_t)(k0 + 2 * half) * 64 + lrow;
            v2f b0v, b1v, b2v, b3v;
            b0v.x = brow[0];  b0v.y = brow[64];
            b1v.x = brow[16]; b1v.y = brow[80];
            b2v.x = brow[32]; b2v.y = brow[96];
            b3v.x = brow[48]; b3v.y = brow[112];
            c0 = __builtin_amdgcn_wmma_f32_16x16x4_f32(false, a, false, b0v, (short)0, c0, false, false);
            c1 = __builtin_amdgcn_wmma_f32_16x16x4_f32(false, a, false, b1v, (short)0, c1, false, false);
            c2 = __builtin_amdgcn_wmma_f32_16x16x4_f32(false, a, false, b2v, (short)0, c2, false, false);
            c3 = __builtin_amdgcn_wmma_f32_16x16x4_f32(false, a, false, b3v, (short)0, c3, false, false);
        }
    }

    // store with relu; D layout: lane<16 -> rows m0+0..7, lane>=16 -> rows m0+8..15
    #pragma unroll
    for (int i = 0; i < 8; ++i) {
        int row = m0 + i + 8 * half;
        if (row < n_nodes) {
            float* o = xout + (size_t)row * 64 + lrow;
            o[0]  = fmaxf(c0[i], 0.0f);
            o[16] = fmaxf(c1[i], 0.0f);
            o[32] = fmaxf(c2[i], 0.0f);
            o[48] = fmaxf(c3[i], 0.0f);
        }
    }
}

// ---------------------------------------------------------------------------
// 5) global mean pool (scatter part)
// ---------------------------------------------------------------------------
__global__ __launch_bounds__(256)
void k_pool(const float* __restrict__ x2,     // [N,64]
            const int* __restrict__ batch,    // [N]
            float* __restrict__ pooled,       // [G,64]
            float* __restrict__ gcnt,         // [G]
            int n_nodes) {
    int idx = blockIdx.x * blockDim.x + threadIdx.x;
    int node = idx >> 6;
    int ch   = idx & 63;
    if (node >= n_nodes) return;
    int g = batch[node];
    atomicAdd(&pooled[(size_t)g * 64 + ch], x2[(size_t)node * 64 + ch]);
    if (ch == 0) atomicAdd(&gcnt[g], 1.0f);
}

// ---------------------------------------------------------------------------
// 6) classifier: out[g,k] = (pooled[g]/max(cnt,1)) @ cls_w + cls_b
// ---------------------------------------------------------------------------
__global__ __launch_bounds__(256)
void k_cls(const float* __restrict__ pooled,  // [G,64]
           const float* __restrict__ gcnt,    // [G]
           const float* __restrict__ cls_w,   // [64,10]
           const float* __restrict__ cls_b,   // [10]
           float* __restrict__ out,           // [G,10]
           int n_graphs) {
    int idx = blockIdx.x * blockDim.x + threadIdx.x;
    int g = idx / 10;
    int k = idx % 10;
    if (g >= n_graphs) return;
    float inv = 1.0f / fmaxf(gcnt[g], 1.0f);
    float acc = cls_b[k];
    #pragma unroll
    for (int f = 0; f < 64; ++f)
        acc += pooled[(size_t)g * 64 + f] * inv * cls_w[f * 10 + k];
    out[(size_t)g * 10 + k] = acc;
}

// ---------------------------------------------------------------------------
// launch
// ---------------------------------------------------------------------------
extern "C" void kernel_launch(void* const* d_in, const int* in_sizes, int n_in,
                              void* d_out, int out_size, void* d_ws, size_t ws_size,
                              hipStream_t stream) {
    const int*   node_feats = (const int*)d_in[0];   // [N,2]
    const int*   edge_index = (const int*)d_in[1];   // [2,E]
    const int*   edge_type  = (const int*)d_in[2];   // [E]
    const int*   batch      = (const int*)d_in[3];   // [N]
    const float* shape_emb  = (const float*)d_in[4];
    const float* color_emb  = (const float*)d_in[5];
    const float* pre_w      = (const float*)d_in[6];
    const float* pre_b      = (const float*)d_in[7];
    const float* root1      = (const float*)d_in[8];
    const float* rel1       = (const float*)d_in[9];
    const float* bias1      = (const float*)d_in[10];
    const float* root2      = (const float*)d_in[11];
    const float* rel2       = (const float*)d_in[12];
    const float* bias2      = (const float*)d_in[13];
    const float* cls_w      = (const float*)d_in[14];
    const float* cls_b      = (const float*)d_in[15];
    float* out = (float*)d_out;

    const int N = in_sizes[3];       // batch: [N]
    const int E = in_sizes[2];       // edge_type: [E]
    const int G = out_size / 10;     // output: [G,10]

    // workspace layout (floats), agg..gcnt contiguous so one zero pass covers them
    float* w      = (float*)d_ws;
    float* x0     = w;                              // N*32
    float* x1     = x0 + (size_t)N * 32;            // N*64
    float* x2     = x1 + (size_t)N * 64;            // N*64
    float* agg    = x2 + (size_t)N * 64;            // N*3*64 (used as N*3*32 in layer 1)
    float* cnt    = agg + (size_t)N * 192;          // N*3
    float* pooled = cnt + (size_t)N * 3;            // G*64
    float* gcnt   = pooled + (size_t)G * 64;        // G

    const int TB = 256;
    auto blocks = [](long long work, int tb) { return (unsigned)((work + tb - 1) / tb); };

    // 1) embed + pre-linear
    k_embed_pre<<<blocks((long long)N * 32, TB), TB, 0, stream>>>(
        node_feats, shape_emb, color_emb, pre_w, pre_b, x0, N);

    // 2) zero agg + cnt + pooled + gcnt (contiguous)
    size_t zlen = (size_t)N * 192 + (size_t)N * 3 + (size_t)G * 64 + (size_t)G;
    k_zero<<<2048, TB, 0, stream>>>(agg, zlen);

    // 3) layer-1 scatter (counts computed here, reused for layer 2)
    k_scatter<32, true><<<blocks((long long)E * 32, TB), TB, 0, stream>>>(
        edge_index, edge_type, x0, agg, cnt, E);
    k_norm<32><<<blocks((long long)N * 3 * 32, TB), TB, 0, stream>>>(agg, cnt, N * 3);

    // 4) layer-1 transform (WMMA)
    k_transform<32><<<blocks(((long long)N + 15) / 16, TB / 32), TB, 0, stream>>>(
        x0, agg, root1, rel1, bias1, x1, N);

    // 5) layer-2: re-zero agg only (cnt identical to layer 1)
    k_zero<<<2048, TB, 0, stream>>>(agg, (size_t)N * 192);
    k_scatter<64, false><<<blocks((long long)E * 64, TB), TB, 0, stream>>>(
        edge_index, edge_type, x1, agg, cnt, E);
    k_norm<64><<<blocks((long long)N * 3 * 64, TB), TB, 0, stream>>>(agg, cnt, N * 3);

    // 6) layer-2 transform (WMMA)
    k_transform<64><<<blocks(((long long)N + 15) / 16, TB / 32), TB, 0, stream>>>(
        x1, agg, root2, rel2, bias2, x2, N);

    // 7) pool + classify
    k_pool<<<blocks((long long)N * 64, TB), TB, 0, stream>>>(x2, batch, pooled, gcnt, N);
    k_cls<<<blocks((long long)G * 10, TB), TB, 0, stream>>>(pooled, gcnt, cls_w, cls_b, out, G);
}